// RNNWithMovingAttention_54425825575663
// MI455X (gfx1250) — compile-verified
//
#include <hip/hip_runtime.h>
#include <math.h>

// Problem constants (fixed by the reference module)
#define B_      32
#define E_      256
#define S_      16
#define M_      16
#define G_      32000
#define NSTEP_  128
#define ROWS_   (B_*NSTEP_)        // 4096 = rows of the vocab GEMM
#define NTILES_ (G_/16)            // 2000 N-tiles of 16 vocab columns
#define NCHUNK_ 8                  // split-K over vocab tiles
#define NT_PER_CHUNK_ (NTILES_/NCHUNK_)  // 250
#define RB_ROWS_ 32                // rows of A per workgroup in phase 2
#define P2_     260                // padded LDS pitch (floats) -> conflict-free b64 reads

typedef __attribute__((ext_vector_type(2))) float v2f;
typedef __attribute__((ext_vector_type(8))) float v8f;

// ---------------------------------------------------------------------------
// Phase 1: recurrent scan. One workgroup per batch item, 256 threads (8 waves).
// f32 throughout (precision of the 128-step recurrence). trans_w streamed from
// L2 each step (4MB, coalesced float4), everything else LDS-resident.
// ---------------------------------------------------------------------------
__global__ __launch_bounds__(256) void phase1_recurrence(
    const float* __restrict__ latent, const float* __restrict__ trans_w,
    const float* __restrict__ trans_b, const float* __restrict__ anchor_w,
    const float* __restrict__ anchor_b, const int* __restrict__ zi,
    float* __restrict__ zs)
{
  __shared__ float xk[S_*E_];        // 16KB  (z0[:,0] * 100)
  __shared__ float xv[S_*E_];        // 16KB  (z0[:,1] * 100)
  __shared__ float aw[M_*E_];        // 16KB  (anchor_w cached)
  __shared__ float ab[M_];
  __shared__ float zbuf[E_];
  __shared__ float znbuf[E_];
  __shared__ float dzpart[8*E_];     // 8KB: per-wave partial dz (deterministic)
  __shared__ float attlog[S_];
  __shared__ float alog[M_];
  __shared__ float att2v[M_];
  __shared__ float wsum[8], wsq[8];

  const int t    = threadIdx.x;
  const int lane = t & 31;
  const int wv   = t >> 5;
  const int b    = blockIdx.x;

  const float* base = latent + (size_t)zi[b] * (2*S_*E_);
  for (int i = t; i < S_*E_; i += 256) {
    xk[i] = base[i] * 100.0f;
    xv[i] = base[S_*E_ + i] * 100.0f;
    aw[i] = anchor_w[i];
  }
  if (t < M_) ab[t] = anchor_b[t];
  zbuf[t] = base[t];                 // z init = z0[:,0,0,:] (unscaled)
  __syncthreads();

  const int sidx = t >> 4;           // 0..15 : attention row / anchor row
  const int l16  = t & 15;           // k-slice within the 16-lane group

  for (int step = 0; step < NSTEP_; ++step) {
    // ---- zn = z / (1e-5 + std(z, ddof=1)) * 0.113 ----
    float zv = zbuf[t];
    float s1 = zv, s2 = zv * zv;
    #pragma unroll
    for (int off = 16; off; off >>= 1) {
      s1 += __shfl_xor(s1, off, 32);
      s2 += __shfl_xor(s2, off, 32);
    }
    if (lane == 0) { wsum[wv] = s1; wsq[wv] = s2; }
    __syncthreads();
    float S1 = 0.f, S2 = 0.f;
    #pragma unroll
    for (int j = 0; j < 8; ++j) { S1 += wsum[j]; S2 += wsq[j]; }
    float mean = S1 * (1.0f / E_);
    float var  = (S2 - S1 * mean) * (1.0f / (E_ - 1));
    float sd   = sqrtf(fmaxf(var, 0.0f));
    float zn_t = zv / (1e-5f + sd) * 0.113f;
    znbuf[t] = zn_t;
    __syncthreads();

    // ---- attention logits: attlog[s] = zn . xk[s] ----
    float p = 0.f;
    #pragma unroll
    for (int j = 0; j < 16; ++j) {
      int k = l16 + (j << 4);
      p = fmaf(znbuf[k], xk[sidx*E_ + k], p);
    }
    #pragma unroll
    for (int off = 8; off; off >>= 1) p += __shfl_xor(p, off, 16);
    if (l16 == 0) attlog[sidx] = p;

    // ---- anchor logits: alog[m] = zn . aw[m] + ab[m] ----
    float q = 0.f;
    #pragma unroll
    for (int j = 0; j < 16; ++j) {
      int k = l16 + (j << 4);
      q = fmaf(znbuf[k], aw[sidx*E_ + k], q);
    }
    #pragma unroll
    for (int off = 8; off; off >>= 1) q += __shfl_xor(q, off, 16);
    if (l16 == 0) alog[sidx] = q + ab[sidx];
    __syncthreads();

    // ---- softmax(att) + emit (thread t owns element e = t) ----
    float mx = -__builtin_inff();
    #pragma unroll
    for (int s = 0; s < S_; ++s) mx = fmaxf(mx, attlog[s]);
    float den = 0.f, em = 0.f;
    #pragma unroll
    for (int s = 0; s < S_; ++s) {
      float w = expf(attlog[s] - mx);
      den += w;
      em = fmaf(w, xv[s*E_ + t], em);
    }
    em /= den;
    zs[((size_t)b * NSTEP_ + step) * E_ + t] = em;   // scan output (emit)

    // ---- softmax(att2) -> att2v[16] ----
    if (t < M_) {
      float m2 = -__builtin_inff();
      #pragma unroll
      for (int m = 0; m < M_; ++m) m2 = fmaxf(m2, alog[m]);
      float d2 = 0.f;
      #pragma unroll
      for (int m = 0; m < M_; ++m) d2 += expf(alog[m] - m2);
      att2v[t] = expf(alog[t] - m2) / d2;
    }
    __syncthreads();

    // ---- att2-weighted GEMV: dz[e] += sum_m att2[m]*(trans_w[m*E+e].zn + b) ----
    // wave wv owns rows r in [wv*512, wv*512+512) -> m in {2wv, 2wv+1}
    const float4* tw4 = (const float4*)trans_w;
    const float4* zn4 = (const float4*)znbuf;
    float4 za = zn4[lane];
    float4 zb = zn4[lane + 32];
    for (int i = 0; i < 512; ++i) {
      int r = (wv << 9) + i;
      float4 wa = tw4[(size_t)r * 64 + lane];
      float4 wb = tw4[(size_t)r * 64 + 32 + lane];
      float acc = wa.x * za.x;
      acc = fmaf(wa.y, za.y, acc);
      acc = fmaf(wa.z, za.z, acc);
      acc = fmaf(wa.w, za.w, acc);
      acc = fmaf(wb.x, zb.x, acc);
      acc = fmaf(wb.y, zb.y, acc);
      acc = fmaf(wb.z, zb.z, acc);
      acc = fmaf(wb.w, zb.w, acc);
      #pragma unroll
      for (int off = 16; off; off >>= 1) acc += __shfl_xor(acc, off, 32);
      if (lane == 0) {
        float val = att2v[r >> 8] * (acc + trans_b[r]);
        int e = r & 255;
        if (i < 256) dzpart[(wv << 8) + e] = val;     // m = 2wv  (first pass)
        else         dzpart[(wv << 8) + e] += val;    // m = 2wv+1 (in-order, deterministic)
      }
    }
    __syncthreads();

    // ---- z = zn + (emit + sum_m att2*mixed2) ----
    float dz = em;
    #pragma unroll
    for (int j = 0; j < 8; ++j) dz += dzpart[(j << 8) + t];
    zbuf[t] = zn_t + dz;
    __syncthreads();
  }
}

// ---------------------------------------------------------------------------
// Phase 2: vocab GEMM + online logsumexp using V_WMMA_F32_16X16X4_F32.
// Block = 64 threads (2 waves), owns 32 rows of A (2 M-tiles, wave == M-tile).
// Grid = 128 row-blocks x 8 vocab chunks (split-K over N-tiles, merged later).
// A/B staged in LDS with pitch 260 floats -> conflict-free ds_load_b64.
// ---------------------------------------------------------------------------
__global__ __launch_bounds__(64) void phase2_vocab_wmma(
    const float* __restrict__ zsbuf, const float* __restrict__ vocab_w,
    const float* __restrict__ vocab_b, const int* __restrict__ y,
    float* __restrict__ pmax, float* __restrict__ psum, float* __restrict__ pylog)
{
  extern __shared__ float smem[];
  float* Alds    = smem;                         // RB_ROWS_ * P2_
  float* Blds    = Alds + RB_ROWS_ * P2_;        // 16 * P2_
  float* bias16  = Blds + 16 * P2_;              // 16
  float* ylogLds = bias16 + 16;                  // RB_ROWS_
  int*   yrow    = (int*)(ylogLds + RB_ROWS_);   // RB_ROWS_

  const int t     = threadIdx.x;
  const int lane  = t & 31;
  const int mt    = t >> 5;        // wave id == M-tile (0..1)
  const int half  = lane >> 4;     // A/B: lanes 16-31 carry K=k0+2,k0+3
  const int l16   = lane & 15;
  const int rb    = blockIdx.x >> 3;   // row block (0..127), 32 rows each
  const int chunk = blockIdx.x & 7;    // split-K chunk over vocab N-tiles

  // Stage A (32 rows x 256) from the emit history
  const float4* zs4 = (const float4*)zsbuf;
  for (int i = t; i < RB_ROWS_ * 64; i += 64) {
    int row = i >> 6, k4 = i & 63;
    float4 v = zs4[(size_t)(rb * RB_ROWS_ + row) * 64 + k4];
    *(float4*)&Alds[row * P2_ + (k4 << 2)] = v;
  }
  if (t < RB_ROWS_) {
    yrow[t]    = y[rb * RB_ROWS_ + t];
    ylogLds[t] = -__builtin_inff();
  }

  float runM[8], runS[8];
  #pragma unroll
  for (int j = 0; j < 8; ++j) { runM[j] = -__builtin_inff(); runS[j] = 0.f; }

  // ISA lane layout for f32 WMMA 16x16x4:
  //  A: lanes 0-15 -> M=lane, K=k0,k0+1 ; lanes 16-31 -> M=lane-16, K=k0+2,k0+3
  //  B: lanes 0-15 -> N=lane, K=k0,k0+1 ; lanes 16-31 -> N=lane-16, K=k0+2,k0+3
  //  C/D: VGPR j -> M = j + 8*(lane/16), N = lane%16
  const int aoff = (mt * 16 + l16) * P2_ + 2 * half;
  const int boff = l16 * P2_ + 2 * half;
  const float4* vw4 = (const float4*)vocab_w;

  for (int nt = chunk * NT_PER_CHUNK_; nt < (chunk + 1) * NT_PER_CHUNK_; ++nt) {
    __syncthreads();   // protect Blds reuse from previous tile
    for (int i = t; i < 16 * 64; i += 64) {
      int n = i >> 6, k4 = i & 63;
      float4 v = vw4[(size_t)(nt * 16 + n) * 64 + k4];
      *(float4*)&Blds[n * P2_ + (k4 << 2)] = v;
    }
    if (t < 16) bias16[t] = vocab_b[nt * 16 + t];
    __syncthreads();

    v8f acc = {};
    #pragma unroll
    for (int k0 = 0; k0 < E_; k0 += 4) {
      v2f a    = *(const v2f*)&Alds[aoff + k0];
      v2f bmat = *(const v2f*)&Blds[boff + k0];
      acc = __builtin_amdgcn_wmma_f32_16x16x4_f32(false, a, false, bmat,
                                                  (short)0, acc, false, false);
    }

    // Online logsumexp over this 16-column tile + picked-logit extraction
    float bn   = bias16[l16];
    int   colg = nt * 16 + l16;
    #pragma unroll
    for (int j = 0; j < 8; ++j) {
      float lv = acc[j] + bn;
      float tm = lv;
      #pragma unroll
      for (int off = 8; off; off >>= 1) tm = fmaxf(tm, __shfl_xor(tm, off, 16));
      float mn = fmaxf(runM[j], tm);
      float e  = expf(lv - mn);
      float es = e;
      #pragma unroll
      for (int off = 8; off; off >>= 1) es += __shfl_xor(es, off, 16);
      runS[j] = fmaf(runS[j], expf(runM[j] - mn), es);
      runM[j] = mn;
      int lr = mt * 16 + j + 8 * half;
      if (yrow[lr] == colg) ylogLds[lr] = lv;   // at most one lane/tile matches
    }
  }
  __syncthreads();

  if (l16 == 0) {   // lanes 0 and 16 of each wave write their 8 rows
    #pragma unroll
    for (int j = 0; j < 8; ++j) {
      int lr  = mt * 16 + j + 8 * half;
      int gr  = rb * RB_ROWS_ + lr;
      int idx = chunk * ROWS_ + gr;
      pmax[idx]  = runM[j];
      psum[idx]  = runS[j];
      pylog[idx] = ylogLds[lr];
    }
  }
}

// ---------------------------------------------------------------------------
// Phase 3: merge the 8 split-K partials: yp = logit_y - (M + log(sum))
// ---------------------------------------------------------------------------
__global__ __launch_bounds__(256) void phase3_merge(
    const float* __restrict__ pmax, const float* __restrict__ psum,
    const float* __restrict__ pylog, float* __restrict__ out)
{
  int r = blockIdx.x * blockDim.x + threadIdx.x;
  if (r >= ROWS_) return;
  float M = -__builtin_inff();
  #pragma unroll
  for (int c = 0; c < NCHUNK_; ++c) M = fmaxf(M, pmax[c * ROWS_ + r]);
  float S = 0.f;
  #pragma unroll
  for (int c = 0; c < NCHUNK_; ++c) S += psum[c * ROWS_ + r] * expf(pmax[c * ROWS_ + r] - M);
  float yl = -__builtin_inff();
  #pragma unroll
  for (int c = 0; c < NCHUNK_; ++c) yl = fmaxf(yl, pylog[c * ROWS_ + r]);
  out[r] = yl - (M + logf(S));
}

// ---------------------------------------------------------------------------
extern "C" void kernel_launch(void* const* d_in, const int* in_sizes, int n_in,
                              void* d_out, int out_size, void* d_ws, size_t ws_size,
                              hipStream_t stream)
{
  (void)in_sizes; (void)n_in; (void)out_size; (void)ws_size;
  const float* latent   = (const float*)d_in[0];
  const float* trans_w  = (const float*)d_in[1];
  const float* trans_b  = (const float*)d_in[2];
  const float* anchor_w = (const float*)d_in[3];
  const float* anchor_b = (const float*)d_in[4];
  const float* vocab_w  = (const float*)d_in[5];
  const float* vocab_b  = (const float*)d_in[6];
  const int*   zi       = (const int*)d_in[7];
  const int*   yidx     = (const int*)d_in[8];
  float* out = (float*)d_out;

  float* ws    = (float*)d_ws;
  float* zs    = ws;                                  // B*NSTEP*E emit history (4MB)
  float* pmax  = zs   + (size_t)B_ * NSTEP_ * E_;     // 8*4096
  float* psum  = pmax + (size_t)NCHUNK_ * ROWS_;      // 8*4096
  float* pylog = psum + (size_t)NCHUNK_ * ROWS_;      // 8*4096

  phase1_recurrence<<<dim3(B_), dim3(256), 0, stream>>>(
      latent, trans_w, trans_b, anchor_w, anchor_b, zi, zs);

  size_t sm2 = ((size_t)RB_ROWS_ * P2_ + 16 * P2_ + 16 + RB_ROWS_) * sizeof(float)
             + RB_ROWS_ * sizeof(int);                // ~50 KB
  phase2_vocab_wmma<<<dim3((ROWS_ / RB_ROWS_) * NCHUNK_), dim3(64), sm2, stream>>>(
      zs, vocab_w, vocab_b, yidx, pmax, psum, pylog);

  phase3_merge<<<dim3(ROWS_ / 256), dim3(256), 0, stream>>>(pmax, psum, pylog, out);
}